// QuantumPoolingLayer_18769007083810
// MI455X (gfx1250) — compile-verified
//
#include <hip/hip_runtime.h>
#include <hip/hip_bf16.h>

typedef __attribute__((ext_vector_type(2))) float v2f;
typedef __attribute__((ext_vector_type(8))) float v8f;

#define CH      512
#define NSLICE  8      // 32 rows / 4 rows-per-slice
#define ROWS    4      // rows per slice
#define TILEF   8192   // floats per tile: 16 cols * 512 ch
#define TILEB   32768  // bytes per tile

// ---------------------------------------------------------------------------
// Kernel 1: stream 256MB input through LDS with async global->LDS copies,
// reduce 16-column half-rows (== one quadrant column) with WMMA ones-matmul.
// Each block: (b, s) slice of 4 rows. Writes per-channel partial sums for
// qj=0/1 to ws. 1024 blocks x 512 threads.
// ---------------------------------------------------------------------------
__global__ __launch_bounds__(512) void qpool_stream_kernel(
    const float* __restrict__ x, float* __restrict__ partial)
{
  extern __shared__ float sbuf[];  // 2 tiles * 8192 floats = 64 KB, LDS offset 0

  const int bs = blockIdx.x;         // 0..1023
  const int b  = bs >> 3;
  const int s  = bs & 7;
  const int tid  = threadIdx.x;
  const int wave = tid >> 5;
  const int lane = tid & 31;
  const int lhalf = lane >> 4;
  const int ln    = lane & 15;

  // accumulators: [channel-group g][qj], all 16 rows of C identical
  v8f acc00 = {}; v8f acc01 = {}; v8f acc10 = {}; v8f acc11 = {};
  const v2f ones = {1.0f, 1.0f};

  // issue async copies for tile t (t = r*2 + h, h == qj column half)
  auto issue_tile = [&](int t) {
    const int r = t >> 1;
    const int h = t & 1;
    const int i = s * 4 + r;
    const int basef = ((b * 32 + i) * 32 + h * 16) * CH;  // contiguous 8192 floats
    const unsigned ldsbase = (unsigned)((t & 1) * TILEB);
#pragma unroll
    for (int k = 0; k < 4; ++k) {
      const unsigned e  = (unsigned)(k * 512 + tid);       // 0..2047 float4 elems
      const unsigned la = ldsbase + e * 16u;
      const unsigned long long ga =
          (unsigned long long)(const float*)(x + basef + e * 4);
      asm volatile("global_load_async_to_lds_b128 %0, %1, off"
                   :: "v"(la), "v"(ga) : "memory");
    }
  };

  issue_tile(0);

#pragma unroll
  for (int t = 0; t < 8; ++t) {
    if (t < 7) {
      issue_tile(t + 1);
      asm volatile("s_wait_asynccnt 4" ::: "memory");  // tile t complete
    } else {
      asm volatile("s_wait_asynccnt 0" ::: "memory");
    }
    __syncthreads();

    // consume tile t: 16 cols x 512 ch; wave handles 32 channels (2 groups)
    const float* tb = sbuf + (t & 1) * TILEF;
    const int qj = t & 1;
#pragma unroll
    for (int g = 0; g < 2; ++g) {
      const int ch = wave * 32 + g * 16 + ln;
#pragma unroll
      for (int jc = 0; jc < 4; ++jc) {
        const int lc = jc * 4 + 2 * lhalf;   // lanes 0-15: K=0,1  lanes 16-31: K=2,3
        v2f bv;
        bv.x = tb[lc * CH + ch];
        bv.y = tb[(lc + 1) * CH + ch];
        v8f* a = (g == 0) ? (qj == 0 ? &acc00 : &acc01)
                          : (qj == 0 ? &acc10 : &acc11);
        *a = __builtin_amdgcn_wmma_f32_16x16x4_f32(
            false, ones, false, bv, (short)0, *a, false, false);
      }
    }
    __syncthreads();
  }

  // every lane holds the column sum for channel (group base + lane&15) in elem 0
  if (lane < 16) {
    const int ch0 = wave * 32 + ln;        // g = 0
    const int ch1 = wave * 32 + 16 + ln;   // g = 1
    float* p0 = partial + ((size_t)bs * CH + ch0) * 2;
    float* p1 = partial + ((size_t)bs * CH + ch1) * 2;
    p0[0] = acc00[0]; p0[1] = acc01[0];
    p1[0] = acc10[0]; p1[1] = acc11[0];
  }
}

// ---------------------------------------------------------------------------
// Kernel 2: combine partials -> quadrant means -> tanh angles -> 4-qubit sim.
// ---------------------------------------------------------------------------
__device__ __forceinline__ void apply1q(
    float sr[16], float si[16], int bp,
    float g00r, float g00i, float g01r, float g01i,
    float g10r, float g10i, float g11r, float g11i)
{
  const int stride = 1 << bp;
#pragma unroll
  for (int base = 0; base < 16; ++base) {
    if (base & stride) continue;
    const int i0 = base, i1 = base + stride;
    const float ar = sr[i0], ai = si[i0], br = sr[i1], bi = si[i1];
    sr[i0] = g00r * ar - g00i * ai + g01r * br - g01i * bi;
    si[i0] = g00r * ai + g00i * ar + g01r * bi + g01i * br;
    sr[i1] = g10r * ar - g10i * ai + g11r * br - g11i * bi;
    si[i1] = g10r * ai + g10i * ar + g11r * bi + g11i * br;
  }
}

__device__ __forceinline__ void cnot_sw(float sr[16], float si[16], int cb, int tb)
{
#pragma unroll
  for (int idx = 0; idx < 16; ++idx) {
    if ((idx & (1 << cb)) && !(idx & (1 << tb))) {
      const int j = idx | (1 << tb);
      float t;
      t = sr[idx]; sr[idx] = sr[j]; sr[j] = t;
      t = si[idx]; si[idx] = si[j]; si[j] = t;
    }
  }
}

__global__ __launch_bounds__(256) void qsim_kernel(
    const float* __restrict__ partial, const float* __restrict__ qw,
    float* __restrict__ out)
{
  const int tid = blockIdx.x * 256 + threadIdx.x;  // 0..65535
  const int b = tid >> 9;
  const int c = tid & 511;

  float q[4] = {0.f, 0.f, 0.f, 0.f};  // k = qi*2 + qj
#pragma unroll
  for (int s = 0; s < 8; ++s) {
    const int qi = s >> 2;
    const float* p = partial + ((size_t)((b << 3) + s) * CH + c) * 2;
    q[qi * 2 + 0] += p[0];
    q[qi * 2 + 1] += p[1];
  }

  const float PI = 3.14159265358979323846f;
  float sr[16], si[16];
#pragma unroll
  for (int i = 0; i < 16; ++i) { sr[i] = 0.f; si[i] = 0.f; }
  sr[0] = 1.f;

  // RY encoding: wire k acts on bit 3-k
#pragma unroll
  for (int k = 0; k < 4; ++k) {
    const float ang = tanhf(q[k] * (1.0f / 256.0f)) * PI;
    const float ch = cosf(ang * 0.5f), sh = sinf(ang * 0.5f);
    apply1q(sr, si, 3 - k, ch, 0.f, -sh, 0.f, sh, 0.f, ch, 0.f);
  }

#pragma unroll
  for (int layer = 0; layer < 2; ++layer) {
#pragma unroll
    for (int k = 0; k < 4; ++k) {
      const float* w = qw + ((size_t)(c * 2 + layer) * 4 + k) * 3;
      const float phi = w[0], th = w[1], om = w[2];
      const float cth = cosf(th * 0.5f), sth = sinf(th * 0.5f);
      const float a = 0.5f * (phi + om), bb = 0.5f * (phi - om);
      const float ca = cosf(a), sa = sinf(a), cb = cosf(bb), sb = sinf(bb);
      // g00=ep*c, g01=-em*s, g10=conj(em)*s, g11=conj(ep)*c
      apply1q(sr, si, 3 - k,
              cth * ca, -cth * sa,
              -sth * cb, -sth * sb,
              sth * cb, -sth * sb,
              cth * ca, cth * sa);
    }
#pragma unroll
    for (int i = 0; i < 4; ++i)
#pragma unroll
      for (int j = i + 1; j < 4; ++j)
        cnot_sw(sr, si, 3 - i, 3 - j);
  }

  float z = 0.f;
#pragma unroll
  for (int idx = 0; idx < 8; ++idx)  z += sr[idx] * sr[idx] + si[idx] * si[idx];
#pragma unroll
  for (int idx = 8; idx < 16; ++idx) z -= sr[idx] * sr[idx] + si[idx] * si[idx];

  out[tid] = z;
}

// ---------------------------------------------------------------------------
extern "C" void kernel_launch(void* const* d_in, const int* in_sizes, int n_in,
                              void* d_out, int out_size, void* d_ws, size_t ws_size,
                              hipStream_t stream)
{
  const float* x  = (const float*)d_in[0];  // (128,32,32,512) f32
  const float* qw = (const float*)d_in[1];  // (512,2,4,3) f32
  float* out = (float*)d_out;               // (128,512) f32
  float* partial = (float*)d_ws;            // 1024*512*2 floats = 4 MB

  qpool_stream_kernel<<<dim3(128 * NSLICE), dim3(512), 2 * TILEB, stream>>>(x, partial);
  qsim_kernel<<<dim3(65536 / 256), dim3(256), 0, stream>>>(partial, qw, out);
}